// SemiSymbolic_21912923144500
// MI455X (gfx1250) — compile-verified
//
#include <hip/hip_runtime.h>

typedef float v2f __attribute__((ext_vector_type(2)));
typedef float v8f __attribute__((ext_vector_type(8)));

#define B_DIM 128
#define N_DIM 1024
#define K_DIM 1024
#define NSPLIT 4            // K split across the 4 waves of a workgroup
#define KSLICE (K_DIM / NSPLIT)   // 256 per wave
#define KC 64               // K chunk staged in LDS per iteration
#define XS_STRIDE 260       // 256 + 4 pad: 16B-aligned rows, rotated bank mapping
#define DELTA 0.01f

__global__ __launch_bounds__(128) void semisym_wmma_splitk_kernel(
    const float* __restrict__ x,   // (B, K) row-major
    const float* __restrict__ W,   // (N, K) row-major
    float* __restrict__ out)       // (B, N)
{
    // 16 rows x (4 strips of 64) each; reused as reduction buffer at the end
    __shared__ float xs[16 * XS_STRIDE];   // 16640 B
    __shared__ float ws[16 * XS_STRIDE];   // 16640 B

    const int tid  = threadIdx.x;          // 0..127
    const int wave = tid >> 5;             // 0..3 (wave32)
    const int lane = tid & 31;
    const int half = lane >> 4;            // lane group 0/1
    const int lm   = lane & 15;

    const int M0 = blockIdx.y * 16;        // batch rows
    const int N0 = blockIdx.x * 16;        // output features

    v8f acc_out = {};
    v8f acc_sum = {};
    float mx[8];
#pragma unroll
    for (int r = 0; r < 8; ++r) mx[r] = 0.0f;

    const int soff = wave * KC;            // this wave's strip inside the LDS tile

    for (int c = 0; c < KSLICE / KC; ++c) {
        // ---- cooperative stage: 4 disjoint K strips (one per wave) ----
        // xs: 16 rows x 256 cols = 1024 float4; 8 per thread. Same for ws.
#pragma unroll
        for (int t = 0; t < 8; ++t) {
            int e   = tid + t * 128;           // 0..1023
            int s   = e >> 8;                  // strip / owning wave
            int f   = e & 255;
            int row = f >> 4;
            int c4  = (f & 15) << 2;
            int gk  = s * KSLICE + c * KC + c4;   // global k
            float4 vx = *(const float4*)&x[(size_t)(M0 + row) * K_DIM + gk];
            *(float4*)&xs[row * XS_STRIDE + s * KC + c4] = vx;
            float4 vw = *(const float4*)&W[(size_t)(N0 + row) * K_DIM + gk];
            *(float4*)&ws[row * XS_STRIDE + s * KC + c4] = vw;
        }
        __syncthreads();

        // ---- WMMA f32 16x16x4: out += A*B ; sum += |A|*|B| ----
        // A frag: lane l -> m = l%16, k = kk + (l/16)*2 (v2f = 2 consecutive k)
        const float* xrow = &xs[lm * XS_STRIDE + soff + half * 2];
        const float* wrow = &ws[lm * XS_STRIDE + soff + half * 2];
#pragma unroll
        for (int kk = 0; kk < KC; kk += 4) {
            v2f a = *(const v2f*)&xrow[kk];
            v2f b = *(const v2f*)&wrow[kk];
            acc_out = __builtin_amdgcn_wmma_f32_16x16x4_f32(
                false, a, false, b, (short)0, acc_out, false, false);
            v2f aa = { __builtin_fabsf(a.x), __builtin_fabsf(a.y) };
            v2f ab = { __builtin_fabsf(b.x), __builtin_fabsf(b.y) };
            acc_sum = __builtin_amdgcn_wmma_f32_16x16x4_f32(
                false, aa, false, ab, (short)0, acc_sum, false, false);
        }

        // ---- tropical part: mx[r] = max_k |x[m][k]| * |w[n][k]| ----
        // per-lane accumulators mirror the WMMA C layout:
        //   lane owns n = lm, rows m = half*8 + r
        {
            const float* wl = &ws[lm * XS_STRIDE + soff];
#pragma unroll 4
            for (int kk = 0; kk < KC; kk += 4) {
                float4 wv = *(const float4*)&wl[kk];
                float aw0 = __builtin_fabsf(wv.x);
                float aw1 = __builtin_fabsf(wv.y);
                float aw2 = __builtin_fabsf(wv.z);
                float aw3 = __builtin_fabsf(wv.w);
#pragma unroll
                for (int r = 0; r < 8; ++r) {
                    float4 xv = *(const float4*)&xs[(half * 8 + r) * XS_STRIDE + soff + kk];
                    float m0 = fmaxf(__builtin_fabsf(xv.x) * aw0,
                                     __builtin_fabsf(xv.y) * aw1);
                    float m1 = fmaxf(__builtin_fabsf(xv.z) * aw2,
                                     __builtin_fabsf(xv.w) * aw3);
                    mx[r] = fmaxf(mx[r], fmaxf(m0, m1));
                }
            }
        }
        __syncthreads();   // all waves done reading before next stage overwrites
    }

    // ---- split-K combine through LDS (reuse xs: 4*32*24 = 3072 floats) ----
    {
        float* red = xs;
        const int base = (wave * 32 + lane) * 24;
#pragma unroll
        for (int r = 0; r < 8; ++r) {
            red[base + r]      = acc_out[r];
            red[base + 8 + r]  = acc_sum[r];
            red[base + 16 + r] = mx[r];
        }
        __syncthreads();

        // 256 output elements; each of the 128 threads finalizes 2
#pragma unroll
        for (int e = tid; e < 256; e += 128) {
            int l = e & 31;        // original lane that held the element
            int r = e >> 5;        // original C VGPR index
            float so = 0.0f, ss = 0.0f, sm = 0.0f;
#pragma unroll
            for (int w = 0; w < NSPLIT; ++w) {
                int idx = (w * 32 + l) * 24 + r;
                so += red[idx];
                ss += red[idx + 8];
                sm  = fmaxf(sm, red[idx + 16]);
            }
            int row = M0 + (l >> 4) * 8 + r;
            int col = N0 + (l & 15);
            out[(size_t)row * N_DIM + col] = so + DELTA * (sm - ss);
        }
    }
}

extern "C" void kernel_launch(void* const* d_in, const int* in_sizes, int n_in,
                              void* d_out, int out_size, void* d_ws, size_t ws_size,
                              hipStream_t stream) {
    (void)in_sizes; (void)n_in; (void)d_ws; (void)ws_size; (void)out_size;
    const float* x = (const float*)d_in[0];   // (128, 1024)
    const float* W = (const float*)d_in[1];   // (1024, 1024)
    float* out = (float*)d_out;               // (128, 1024)

    dim3 grid(N_DIM / 16, B_DIM / 16, 1);     // 64 x 8 = 512 workgroups
    dim3 block(128, 1, 1);                    // 4 wave32 waves, split-K
    semisym_wmma_splitk_kernel<<<grid, block, 0, stream>>>(x, W, out);
}